// SPConvDown_49924699849228
// MI455X (gfx1250) — compile-verified
//
#include <hip/hip_runtime.h>
#include <hip/hip_bf16.h>

#define N_IN_C   2000000
#define N_OUT_C  250000
#define C_IN     32
#define C_OUT    64
#define K_VOL    8
#define EPS_F    1e-5f

// Flip to 0 if the gfx1250 assembler rejects the async-to-LDS mnemonic.
#define USE_ASYNC_LDS 1

typedef __attribute__((ext_vector_type(2))) float v2f;
typedef __attribute__((ext_vector_type(8))) float v8f;

// ---------------- workspace layout (bytes) ----------------
// [0..32)    counts[8]
// [32..64)   offsets[8]
// [64..96)   cursors[8]
// [128..640) sums: 64 x f32 sum, 64 x f32 sumsq
// [640..1152) params: A[64], B[64]
// [4096.. )  sorted point indices: N_IN ints (8 MB)

__global__ void zero_f32_kernel(float* p, int n) {
    int i = blockIdx.x * blockDim.x + threadIdx.x;
    for (; i < n; i += gridDim.x * blockDim.x) p[i] = 0.0f;
}

__global__ void count_kernel(const int* __restrict__ offid, int* __restrict__ counts, int n) {
    int i = blockIdx.x * blockDim.x + threadIdx.x;
    for (; i < n; i += gridDim.x * blockDim.x)
        atomicAdd(&counts[offid[i]], 1);
}

__global__ void scan_kernel(const int* __restrict__ counts, int* __restrict__ offsets,
                            int* __restrict__ cursors) {
    if (threadIdx.x == 0 && blockIdx.x == 0) {
        int acc = 0;
        for (int k = 0; k < K_VOL; ++k) {
            offsets[k] = acc;
            cursors[k] = acc;
            acc += counts[k];
        }
    }
}

__global__ void scatter_kernel(const int* __restrict__ offid, int* __restrict__ cursors,
                               int* __restrict__ sorted, int n) {
    int i = blockIdx.x * blockDim.x + threadIdx.x;
    for (; i < n; i += gridDim.x * blockDim.x) {
        int pos = atomicAdd(&cursors[offid[i]], 1);
        sorted[pos] = i;
    }
}

// ---------------- WMMA gather-GEMM-scatter conv (persistent waves) ----------------
// Wave32 per 16-row tile of the tap-sorted point list, grid-stride over tiles.
// D(16x64) = A(16x32) @ weight[k](32x64) via 8 K-steps x 4 N-tiles of
// v_wmma_f32_16x16x4_f32, then atomic scatter-add rows into out[out_idx].
__global__ __launch_bounds__(256) void spconv_wmma_kernel(
    const float* __restrict__ feats, const int* __restrict__ offid,
    const int* __restrict__ oidx, const float* __restrict__ weight,
    const int* __restrict__ sorted, float* __restrict__ out,
    int nIn, int nTiles)
{
    __shared__ float w_lds[K_VOL * C_IN * C_OUT];  // 64 KB (WGP LDS is 320 KB)

#if USE_ASYNC_LDS
    // Async DMA of all 8 taps' weights into LDS (GVS mode: VDST = LDS byte
    // address, VADDR = 32-bit global offset, SADDR = base). Tracked by
    // ASYNCcnt. LDS dest derived from the array's actual generic address:
    // low 32 bits of a generic LDS pointer are the LDS byte offset.
    {
        const unsigned ldsBase = (unsigned)(size_t)(void*)&w_lds[0];
        unsigned i = threadIdx.x;
#pragma unroll
        for (int t = 0; t < 16; ++t, i += 256) {
            unsigned goff = i * 16u;
            unsigned loff = ldsBase + goff;
            asm volatile("global_load_async_to_lds_b128 %0, %1, %2"
                         :
                         : "v"(loff), "v"(goff), "s"(weight)
                         : "memory");
        }
        asm volatile("s_wait_asynccnt 0x0" ::: "memory");
    }
#else
    for (int i = threadIdx.x; i < (K_VOL * C_IN * C_OUT) / 4; i += 256)
        ((float4*)w_lds)[i] = ((const float4*)weight)[i];
#endif
    __syncthreads();

    const int wave = threadIdx.x >> 5;
    const int lane = threadIdx.x & 31;
    const int half = lane >> 4;    // 0: K low pair / C rows 0-7; 1: K high pair / C rows 8-15
    const int mrow = lane & 15;    // A row this lane loads; C column-within-ntile
    const int waveStride = gridDim.x * 8;

    for (int tile = blockIdx.x * 8 + wave; tile < nTiles; tile += waveStride) {
        const int r = tile * 16 + mrow;
        const bool valid = (r < nIn);
        const int p    = valid ? sorted[r] : 0;
        const int tapv = valid ? offid[p] : -1;
        const int my_oidx = valid ? oidx[p] : 0;

        // Preload this lane's half of A row `mrow`: K = kb*4 + 2*half + {0,1}.
        // Lanes l and l+16 interleave to cover the full 128B feature row once.
        v2f a_pre[8];
        const float* fr = feats + (size_t)p * C_IN + half * 2;
#pragma unroll
        for (int kb = 0; kb < 8; ++kb) {
            a_pre[kb].x = fr[kb * 4 + 0];
            a_pre[kb].y = fr[kb * 4 + 1];
        }

        v8f acc0 = {}, acc1 = {}, acc2 = {}, acc3 = {};
        const v2f zero2 = {};

        for (int k = 0; k < K_VOL; ++k) {
            if (!__any(tapv == k)) continue;       // wave-uniform; EXEC stays all-1s
            const float* wk = w_lds + k * C_IN * C_OUT;
#pragma unroll
            for (int kb = 0; kb < 8; ++kb) {
                v2f a = (tapv == k) ? a_pre[kb] : zero2;  // mask off-tap rows in registers
                const int krow = kb * 4 + half * 2;
                const float* wb = wk + krow * C_OUT + mrow;
                v2f b0, b1, b2, b3;
                b0.x = wb[0];   b0.y = wb[C_OUT + 0];
                b1.x = wb[16];  b1.y = wb[C_OUT + 16];
                b2.x = wb[32];  b2.y = wb[C_OUT + 32];
                b3.x = wb[48];  b3.y = wb[C_OUT + 48];
                acc0 = __builtin_amdgcn_wmma_f32_16x16x4_f32(false, a, false, b0, (short)0, acc0, false, false);
                acc1 = __builtin_amdgcn_wmma_f32_16x16x4_f32(false, a, false, b1, (short)0, acc1, false, false);
                acc2 = __builtin_amdgcn_wmma_f32_16x16x4_f32(false, a, false, b2, (short)0, acc2, false, false);
                acc3 = __builtin_amdgcn_wmma_f32_16x16x4_f32(false, a, false, b3, (short)0, acc3, false, false);
            }
        }

        // Scatter: lane holds C rows m = half*8 + v at column nt*16 + mrow.
        // Row m's out_idx lives in lane m (m < 16): fetch via shuffle.
#pragma unroll
        for (int v = 0; v < 8; ++v) {
            int m = half * 8 + v;
            int orow = __shfl(my_oidx, m, 32);
            float* obase = out + (size_t)orow * C_OUT + mrow;
            atomicAdd(obase + 0,  acc0[v]);
            atomicAdd(obase + 16, acc1[v]);
            atomicAdd(obase + 32, acc2[v]);
            atomicAdd(obase + 48, acc3[v]);
        }
    }
}

// ---------------- BN stats: per-channel sum / sumsq ----------------
__global__ __launch_bounds__(256) void stats_kernel(const float* __restrict__ out,
                                                    float* __restrict__ sums, int nOut) {
    int c = threadIdx.x & (C_OUT - 1);
    int rbase = blockIdx.x * 4 + (threadIdx.x >> 6);
    int rstride = gridDim.x * 4;
    float s = 0.0f, s2 = 0.0f;
    for (int r = rbase; r < nOut; r += rstride) {
        float v = out[(size_t)r * C_OUT + c];
        s += v;
        s2 += v * v;
    }
    atomicAdd(&sums[c], s);
    atomicAdd(&sums[C_OUT + c], s2);
}

// ---------------- fold BN + condition affine + context modulation ----------------
__global__ void params_kernel(const float* __restrict__ gamma, const float* __restrict__ beta,
                              const float* __restrict__ context, const float* __restrict__ ctx_w,
                              const int* __restrict__ cond, const float* __restrict__ sums,
                              float* __restrict__ prm, int nOut) {
    int c = threadIdx.x;  // 64 threads
    float sc = 0.0f, sh = 0.0f;
    for (int j = 0; j < 256; ++j) {
        float cj = context[j];
        sc += cj * ctx_w[j * (2 * C_OUT) + c];
        sh += cj * ctx_w[j * (2 * C_OUT) + C_OUT + c];
    }
    int cd = cond[0];
    float inv_n = 1.0f / (float)nOut;
    float mean = sums[c] * inv_n;
    float var  = sums[C_OUT + c] * inv_n - mean * mean;
    float inv  = rsqrtf(var + EPS_F);
    float g = gamma[cd * C_OUT + c];
    float b = beta[cd * C_OUT + c];
    float gm = inv * g * (1.0f + sc);
    float bm = (b - mean * inv * g) * (1.0f + sc) + sh;
    prm[c] = gm;
    prm[C_OUT + c] = bm;
}

__global__ void finalize_kernel(float* __restrict__ out, const float* __restrict__ prm, int n) {
    int i = blockIdx.x * blockDim.x + threadIdx.x;
    for (; i < n; i += gridDim.x * blockDim.x) {
        int c = i & (C_OUT - 1);
        float v = out[i] * prm[c] + prm[C_OUT + c];
        out[i] = v > 0.0f ? v : 0.0f;
    }
}

extern "C" void kernel_launch(void* const* d_in, const int* in_sizes, int n_in,
                              void* d_out, int out_size, void* d_ws, size_t ws_size,
                              hipStream_t stream) {
    const float* feats   = (const float*)d_in[0];
    const int*   offid   = (const int*)d_in[1];
    const int*   oidx    = (const int*)d_in[2];
    const float* weight  = (const float*)d_in[3];
    const float* gamma   = (const float*)d_in[4];
    const float* beta    = (const float*)d_in[5];
    const float* context = (const float*)d_in[6];
    const float* ctx_w   = (const float*)d_in[7];
    const int*   cond    = (const int*)d_in[8];

    float* out = (float*)d_out;
    char*  ws  = (char*)d_ws;

    int*   counts  = (int*)(ws + 0);
    int*   offsets = (int*)(ws + 32);
    int*   cursors = (int*)(ws + 64);
    float* sums    = (float*)(ws + 128);   // 128 floats
    float* prm     = (float*)(ws + 640);   // 128 floats
    int*   sorted  = (int*)(ws + 4096);    // N_IN ints

    const int nIn  = in_sizes[1];          // 2,000,000
    const int nOut = N_OUT_C;
    const int nElems = nOut * C_OUT;       // 16,000,000

    // 1. zero output accumulator + ws header (counts..params region)
    zero_f32_kernel<<<4096, 256, 0, stream>>>(out, nElems);
    zero_f32_kernel<<<1, 256, 0, stream>>>((float*)ws, 288);  // first 1152 bytes

    // 2. counting sort by kernel tap
    count_kernel<<<2048, 256, 0, stream>>>(offid, counts, nIn);
    scan_kernel<<<1, 32, 0, stream>>>(counts, offsets, cursors);
    scatter_kernel<<<2048, 256, 0, stream>>>(offid, cursors, sorted, nIn);

    // 3. WMMA gather-GEMM-scatter conv: persistent blocks, grid-stride tiles
    const int nTiles = (nIn + 15) / 16;    // 125,000
    const int nBlocks = 2048;              // 16,384 waves; ~8 tiles per wave
    spconv_wmma_kernel<<<nBlocks, 256, 0, stream>>>(feats, offid, oidx, weight,
                                                    sorted, out, nIn, nTiles);

    // 4. BN stats, fused affine params, in-place ReLU affine
    stats_kernel<<<1024, 256, 0, stream>>>(out, sums, nOut);
    params_kernel<<<1, 64, 0, stream>>>(gamma, beta, context, ctx_w, cond, sums, prm, nOut);
    finalize_kernel<<<8192, 256, 0, stream>>>(out, prm, nElems);
}